// CombinedLoss_17643725652266
// MI455X (gfx1250) — compile-verified
//
#include <hip/hip_runtime.h>

typedef float v2f __attribute__((ext_vector_type(2)));
typedef float v4f __attribute__((ext_vector_type(4)));
typedef float v8f __attribute__((ext_vector_type(8)));

#define NACC 7
// acc slots: 0=robot_sq 1=phase_nll 2=coh_sum 3=coh_cnt 4=speed_pen 5=vel_sq 6=acc_sq

// ---------------- wave32 reduction: WMMA-based if available ----------------
__device__ __forceinline__ float wave_sum(float v) {
#if __has_builtin(__builtin_amdgcn_wmma_f32_16x16x4_f32)
  // A (16x4 f32, 2 VGPRs/lane): lanes 0-15 hold K=0,1 ; lanes 16-31 hold K=2,3.
  // a = {v, 0}  ->  A[m,0]=v_m, A[m,2]=v_{m+16}, cols 1,3 = 0.
  // B (4x16) = all ones -> D[m,n] = v_m + v_{m+16} (replicated over n).
  v2f a; a[0] = v;   a[1] = 0.0f;
  v2f b; b[0] = 1.0f; b[1] = 1.0f;
  v8f c = {};
  v8f d = __builtin_amdgcn_wmma_f32_16x16x4_f32(false, a, false, b, (short)0, c,
                                                false, false);
  // lanes 0-15: rows 0..7 ; lanes 16-31: rows 8..15 (each replicated per lane)
  float s = d[0] + d[1] + d[2] + d[3] + d[4] + d[5] + d[6] + d[7];
  s += __shfl_xor(s, 16);   // combine the two half-wave row groups
  return s;
#else
  for (int m = 16; m >= 1; m >>= 1) v += __shfl_xor(v, m);
  return v;
#endif
}

__device__ __forceinline__ void load12(const float* p, float* r) {
  v4f a0 = ((const v4f*)p)[0];
  v4f a1 = ((const v4f*)p)[1];
  v4f a2 = ((const v4f*)p)[2];
  r[0]=a0[0]; r[1]=a0[1]; r[2] =a0[2]; r[3] =a0[3];
  r[4]=a1[0]; r[5]=a1[1]; r[6] =a1[2]; r[7] =a1[3];
  r[8]=a2[0]; r[9]=a2[1]; r[10]=a2[2]; r[11]=a2[3];
}

// gt_robot is read exactly once -> non-temporal so reusable data stays in L2
__device__ __forceinline__ void load12_nt(const float* p, float* r) {
  v4f a0 = __builtin_nontemporal_load((const v4f*)p + 0);
  v4f a1 = __builtin_nontemporal_load((const v4f*)p + 1);
  v4f a2 = __builtin_nontemporal_load((const v4f*)p + 2);
  r[0]=a0[0]; r[1]=a0[1]; r[2] =a0[2]; r[3] =a0[3];
  r[4]=a1[0]; r[5]=a1[1]; r[6] =a1[2]; r[7] =a1[3];
  r[8]=a2[0]; r[9]=a2[1]; r[10]=a2[2]; r[11]=a2[3];
}

__global__ void zero_ws_kernel(double* ws) {
  if (threadIdx.x < NACC) ws[threadIdx.x] = 0.0;
}

__global__ __launch_bounds__(256)
void loss_main_kernel(const float* __restrict__ pred_robot,
                      const float* __restrict__ pred_phase,
                      const float* __restrict__ gt_robot,
                      const int*  __restrict__ gt_phase,
                      double* __restrict__ ws,
                      int Bsz, int T) {
  const int K = 8;                 // timesteps per strip (T % K == 0)
  const int S = T / K;             // strips per batch row
  const int n_strips = Bsz * S;

  float acc[NACC];
#pragma unroll
  for (int i = 0; i < NACC; ++i) acc[i] = 0.0f;

  const int stride = gridDim.x * blockDim.x;
  for (int idx = blockIdx.x * blockDim.x + threadIdx.x; idx < n_strips;
       idx += stride) {
    const int b  = idx / S;
    const int t0 = (idx - b * S) * K;
    const size_t rowbase = (size_t)b * (size_t)T;

    float rm1[12], rm2[12];
#pragma unroll
    for (int i = 0; i < 12; ++i) { rm1[i] = 0.0f; rm2[i] = 0.0f; }
    if (t0 > 0) {                  // t0 is a multiple of K>=8 so t0-2 >= 0
      load12(pred_robot + (rowbase + (size_t)(t0 - 1)) * 12, rm1);
      load12(pred_robot + (rowbase + (size_t)(t0 - 2)) * 12, rm2);
    }

    // carried phase logits for timestep t
    float pc0, pc1, pc2;
    {
      const float* pp = pred_phase + (rowbase + (size_t)t0) * 3;
      pc0 = pp[0]; pc1 = pp[1]; pc2 = pp[2];
    }

#pragma unroll
    for (int k = 0; k < K; ++k) {
      const int t = t0 + k;
      float r[12], g[12];
      load12   (pred_robot + (rowbase + (size_t)t) * 12, r);
      load12_nt(gt_robot   + (rowbase + (size_t)t) * 12, g);

      // robot MSE
      float rs = 0.0f;
#pragma unroll
      for (int i = 0; i < 12; ++i) { float d = r[i] - g[i]; rs = fmaf(d, d, rs); }
      acc[0] += rs;

      if (t >= 1) {
        float vel[12];
        float vs = 0.0f;
#pragma unroll
        for (int i = 0; i < 12; ++i) {
          vel[i] = r[i] - rm1[i];
          vs = fmaf(vel[i], vel[i], vs);
        }
        acc[5] += vs;                       // smooth-vel
#pragma unroll
        for (int j = 0; j < 4; ++j) {       // 4 points x 3D speed penalty
          float s2 = vel[3*j]*vel[3*j] + vel[3*j+1]*vel[3*j+1]
                   + vel[3*j+2]*vel[3*j+2];
          float pen = sqrtf(s2) - 10.0f;    // MAX_SPEED
          pen = pen > 0.0f ? pen : 0.0f;
          acc[4] = fmaf(pen, pen, acc[4]);
        }
        if (t >= 2) {
          float as = 0.0f;
#pragma unroll
          for (int i = 0; i < 12; ++i) {
            float a = r[i] - 2.0f * rm1[i] + rm2[i];
            as = fmaf(a, a, as);
          }
          acc[6] += as;                     // smooth-acc
        }
      }

      // phase cross-entropy at t
      float m   = fmaxf(pc0, fmaxf(pc1, pc2));
      float lse = m + logf(expf(pc0 - m) + expf(pc1 - m) + expf(pc2 - m));
      int   gt  = gt_phase[rowbase + (size_t)t];
      float pg  = (gt == 0) ? pc0 : ((gt == 1) ? pc1 : pc2);
      acc[1] += lse - pg;

      // argmax at t (first-max semantics like jnp.argmax)
      int pt = 0; float bt = pc0;
      if (pc1 > bt) { bt = pc1; pt = 1; }
      if (pc2 > bt) { bt = pc2; pt = 2; }

      if (t < T - 1) {
        const float* pp = pred_phase + (rowbase + (size_t)t + 1) * 3;
        float pn0 = pp[0], pn1 = pp[1], pn2 = pp[2];
        int pt1 = 0; float bn = pn0;
        if (pn1 > bn) { bn = pn1; pt1 = 1; }
        if (pn2 > bn) { bn = pn2; pt1 = 2; }
        // illegal transitions (0,2)(1,0)(2,0)(2,1) -> bitmask over pt*3+pt1
        const int code = pt * 3 + pt1;
        if ((0xCC >> code) & 1) {
          acc[2] = fmaf(bn, bn, acc[2]);    // sel = max logit at t+1
          acc[3] += 1.0f;
        }
        pc0 = pn0; pc1 = pn1; pc2 = pn2;    // carry to next iteration
      }

#pragma unroll
      for (int i = 0; i < 12; ++i) { rm2[i] = rm1[i]; rm1[i] = r[i]; }
    }
  }

  // ---- wave reduction (WMMA), then block reduction, then f64 atomics ----
#pragma unroll
  for (int i = 0; i < NACC; ++i) acc[i] = wave_sum(acc[i]);

  __shared__ float sred[8][NACC];
  const int wave = threadIdx.x >> 5;
  const int lane = threadIdx.x & 31;
  if (lane == 0) {
#pragma unroll
    for (int i = 0; i < NACC; ++i) sred[wave][i] = acc[i];
  }
  __syncthreads();
  if (threadIdx.x == 0) {
    const int nwaves = blockDim.x >> 5;
#pragma unroll
    for (int i = 0; i < NACC; ++i) {
      double s = 0.0;
      for (int w = 0; w < nwaves; ++w) s += (double)sred[w][i];
      atomicAdd(&ws[i], s);                 // global_atomic_add_f64
    }
  }
}

__global__ void loss_final_kernel(const double* __restrict__ ws,
                                  float* __restrict__ out, int Bsz, int T) {
  if (threadIdx.x == 0 && blockIdx.x == 0) {
    const double nBT   = (double)Bsz * (double)T;
    double robot = ws[0] / (nBT * 12.0);
    double phase = ws[1] / nBT;
    double coh   = (ws[3] > 0.0) ? (ws[2] / ws[3]) : 0.0;
    double speed = 5.0  * (ws[4] / ((double)Bsz * (double)(T - 1) * 4.0));
    double vel   = 0.05 * (ws[5] / ((double)Bsz * (double)(T - 1) * 12.0));
    double accm  = 0.01 * (ws[6] / ((double)Bsz * (double)(T - 2) * 12.0));
    out[0] = (float)(robot + phase + 10.0 * coh + speed + vel + accm);
  }
}

extern "C" void kernel_launch(void* const* d_in, const int* in_sizes, int n_in,
                              void* d_out, int out_size, void* d_ws,
                              size_t ws_size, hipStream_t stream) {
  const float* pred_robot = (const float*)d_in[0];
  const float* pred_phase = (const float*)d_in[1];
  const float* gt_robot   = (const float*)d_in[2];
  const int*   gt_phase   = (const int*)d_in[3];
  float* out = (float*)d_out;
  double* ws = (double*)d_ws;

  const int T   = 8192;
  const int Bsz = in_sizes[3] / T;   // gt_phase is (B, T)

  zero_ws_kernel<<<1, 32, 0, stream>>>(ws);

  const int K = 8;
  const int n_strips = Bsz * (T / K);
  const int block = 256;
  int grid = (n_strips + block - 1) / block;
  if (grid > 16384) grid = 16384;

  loss_main_kernel<<<grid, block, 0, stream>>>(pred_robot, pred_phase,
                                               gt_robot, gt_phase, ws, Bsz, T);
  loss_final_kernel<<<1, 1, 0, stream>>>(ws, out, Bsz, T);
}